// GNNEncoder_7945689497634
// MI455X (gfx1250) — compile-verified
//
#include <hip/hip_runtime.h>

typedef __attribute__((ext_vector_type(2))) float v2f;
typedef __attribute__((ext_vector_type(8))) float v8f;

#define N_NODES 100000
#define N_EDGES 640000
#define IN_DIM  128
#define OUT_DIM 128

// ---------------------------------------------------------------------------
// 1) degree init: every node starts with its self-loop (deg = 1)
// ---------------------------------------------------------------------------
__global__ void gcn_deg_init(float* __restrict__ deg) {
    int n = blockIdx.x * blockDim.x + threadIdx.x;
    if (n < N_NODES) deg[n] = 1.0f;
}

// ---------------------------------------------------------------------------
// 2) degree accumulation over edge destinations
// ---------------------------------------------------------------------------
__global__ void gcn_deg_edges(const int* __restrict__ dst, float* __restrict__ deg) {
    int e = blockIdx.x * blockDim.x + threadIdx.x;
    if (e < N_EDGES) atomicAdd(&deg[dst[e]], 1.0f);
}

// ---------------------------------------------------------------------------
// 3) dinv = rsqrt(deg) (deg >= 1 always, keep reference's guard anyway)
// ---------------------------------------------------------------------------
__global__ void gcn_dinv(const float* __restrict__ deg, float* __restrict__ dinv) {
    int n = blockIdx.x * blockDim.x + threadIdx.x;
    if (n < N_NODES) {
        float d = deg[n];
        dinv[n] = d > 0.0f ? rsqrtf(d) : 0.0f;
    }
}

// ---------------------------------------------------------------------------
// 4) h = x @ W^T via V_WMMA_F32_16X16X4_F32.
//    One wave -> one 16x16 tile of h. Block = 8 waves = all 8 column tiles
//    of one 16-node row group. Grid = 100000/16 = 6250 blocks.
//
//    A-matrix 16x4 f32 layout:  vgpr j, lane(half,r) holds A[r, 2*half+j]
//    B-matrix 4x16 f32 layout:  vgpr j, lane(half,r) holds B[2*half+j, r]
//      with B = W^T  =>  b[j] = W[o0+r, k0+2*half+j]   (contiguous float2)
//    C/D 16x16 f32:             vgpr v, lane(half,r) holds D[v+8*half, r]
// ---------------------------------------------------------------------------
__global__ void gcn_gemm_wmma(const float* __restrict__ x,
                              const float* __restrict__ W,
                              float* __restrict__ h) {
    const int lane = threadIdx.x & 31;
    const int wave = threadIdx.x >> 5;   // 0..7 -> output column tile
    const int half = lane >> 4;          // 0 or 1
    const int r    = lane & 15;
    const int n0   = blockIdx.x * 16;
    const int o0   = wave * 16;

    const float* arow = x + (size_t)(n0 + r) * IN_DIM + 2 * half;
    const float* brow = W + (size_t)(o0 + r) * IN_DIM + 2 * half;

    v8f c = {};
#pragma unroll
    for (int k0 = 0; k0 < IN_DIM; k0 += 4) {
        v2f a = *(const v2f*)(arow + k0);
        v2f b = *(const v2f*)(brow + k0);
        // 8 args: (neg_a, A, neg_b, B, c_mod, C, reuse_a, reuse_b)
        c = __builtin_amdgcn_wmma_f32_16x16x4_f32(
                /*neg_a=*/false, a, /*neg_b=*/false, b,
                /*c_mod=*/(short)0, c, /*reuse_a=*/false, /*reuse_b=*/false);
    }

#pragma unroll
    for (int v = 0; v < 8; ++v) {
        h[(size_t)(n0 + v + 8 * half) * OUT_DIM + (o0 + r)] = c[v];
    }
}

// ---------------------------------------------------------------------------
// 5) self-loop contribution: out[n,f] = h[n,f] * dinv[n]^2
//    (also serves as the accumulator initialization for the scatter pass)
// ---------------------------------------------------------------------------
__global__ void gcn_self_init(const float* __restrict__ h,
                              const float* __restrict__ dinv,
                              float* __restrict__ out) {
    int t = blockIdx.x * blockDim.x + threadIdx.x;
    if (t < N_NODES * OUT_DIM) {
        int n = t >> 7;
        float di = dinv[n];
        out[t] = h[t] * di * di;
    }
}

// ---------------------------------------------------------------------------
// 6) edge scatter-add: out[dst] += h[src] * dinv[src]*dinv[dst]
//    One thread per (edge, feature). 128 consecutive threads share one edge:
//    index/dinv loads coalesce to single requests, h gather + atomics are
//    fully coalesced 512B bursts. h (51MB) and out (51MB) are L2-resident
//    (192MB L2), so the 327MB logical gather stays on-die.
// ---------------------------------------------------------------------------
__global__ void gcn_scatter(const float* __restrict__ h,
                            const float* __restrict__ dinv,
                            const int* __restrict__ src,
                            const int* __restrict__ dst,
                            float* __restrict__ out) {
    int t = blockIdx.x * blockDim.x + threadIdx.x;
    if (t < N_EDGES * OUT_DIM) {
        int e = t >> 7;
        int f = t & 127;
        int s = src[e];
        int d = dst[e];
        float norm = dinv[s] * dinv[d];
        atomicAdd(&out[(size_t)d * OUT_DIM + f], h[(size_t)s * OUT_DIM + f] * norm);
    }
}

// ---------------------------------------------------------------------------
// 7) bias + ReLU, in place
// ---------------------------------------------------------------------------
__global__ void gcn_finalize(const float* __restrict__ bias, float* __restrict__ out) {
    int t = blockIdx.x * blockDim.x + threadIdx.x;
    if (t < N_NODES * OUT_DIM) {
        float v = out[t] + bias[t & 127];
        out[t] = v > 0.0f ? v : 0.0f;
    }
}

// ---------------------------------------------------------------------------
extern "C" void kernel_launch(void* const* d_in, const int* in_sizes, int n_in,
                              void* d_out, int out_size, void* d_ws, size_t ws_size,
                              hipStream_t stream) {
    const float* x    = (const float*)d_in[0];
    const int*   edge = (const int*)d_in[1];   // [2, E]: row0=src, row1=dst
    const float* W    = (const float*)d_in[2];
    const float* bias = (const float*)d_in[3];
    float* out = (float*)d_out;

    const int* src = edge;
    const int* dst = edge + N_EDGES;

    // workspace layout (all 256B-aligned): h | deg | dinv  (~52 MB total)
    char*  ws   = (char*)d_ws;
    float* h    = (float*)(ws);                               // 100000*128*4 = 51,200,000 B
    float* deg  = (float*)(ws + 51200000);                    // 400,000 B
    float* dinv = (float*)(ws + 51200000 + 401408);           // 400,000 B

    // 1) deg = 1 (self loops)
    gcn_deg_init<<<(N_NODES + 255) / 256, 256, 0, stream>>>(deg);
    // 2) deg[dst] += 1 per edge
    gcn_deg_edges<<<(N_EDGES + 255) / 256, 256, 0, stream>>>(dst, deg);
    // 3) dinv = rsqrt(deg)
    gcn_dinv<<<(N_NODES + 255) / 256, 256, 0, stream>>>(deg, dinv);
    // 4) h = x @ W^T  (fp32 WMMA)
    gcn_gemm_wmma<<<N_NODES / 16, 256, 0, stream>>>(x, W, h);
    // 5) out = h * dinv^2   (self-loop term, initializes accumulator)
    gcn_self_init<<<(N_NODES * OUT_DIM + 255) / 256, 256, 0, stream>>>(h, dinv, out);
    // 6) out[dst] += h[src] * dinv[src]*dinv[dst]
    gcn_scatter<<<(N_EDGES * OUT_DIM + 255) / 256, 256, 0, stream>>>(h, dinv, src, dst, out);
    // 7) out = relu(out + b)
    gcn_finalize<<<(N_NODES * OUT_DIM + 255) / 256, 256, 0, stream>>>(bias, out);
}